// LightGCN_80831284511424
// MI455X (gfx1250) — compile-verified
//
#include <hip/hip_runtime.h>
#include <hip/hip_bf16.h>
#include <math.h>

// ---------------- problem constants (match reference) ----------------
#define N_USERS 100000
#define N_ITEMS 100000
#define EMB     64
#define NNZ     6400000LL
#define BATCH   16384
#define N_NODES (N_USERS + N_ITEMS)

typedef __attribute__((ext_vector_type(2))) float v2f;
typedef __attribute__((ext_vector_type(8))) float v8f;

// ---------------------------------------------------------------------
// SpMM: y[row[e], :] += val[e] * x[col[e], :]   (EMB = 64)
// One wave32 per edge: lane p handles elements {2p, 2p+1} as a float2.
// Edge gather = 256B = 2 full cachelines; scatter via global_atomic_add_f32.
// Source is given as two pointers (user part / item part) so layer 0 can
// read the original embedding tables without a concat copy.
// ---------------------------------------------------------------------
__global__ __launch_bounds__(256) void lightgcn_spmm(
    const int* __restrict__ row, const int* __restrict__ col,
    const float* __restrict__ val,
    const float* __restrict__ srcU, const float* __restrict__ srcI,
    float* __restrict__ dst, long long nnz)
{
    long long g = (long long)blockIdx.x * blockDim.x + threadIdx.x;
    long long e = g >> 5;            // edge index (one wave per edge)
    int       p = (int)(g & 31);     // lane: element pair within the row
    if (e >= nnz) return;

    int   r = row[e];
    int   c = col[e];
    float v = val[e];

    const float* src = (c < N_USERS)
        ? (srcU + (long long)c * EMB)
        : (srcI + (long long)(c - N_USERS) * EMB);

    v2f x = *(const v2f*)(src + 2 * p);            // global_load_b64
    float* y = dst + (long long)r * EMB + 2 * p;
    atomicAdd(y,     v * x.x);                     // global_atomic_add_f32
    atomicAdd(y + 1, v * x.y);
}

// ---------------------------------------------------------------------
// stable softplus
// ---------------------------------------------------------------------
__device__ __forceinline__ float softplusf(float x)
{
    return fmaxf(x, 0.0f) + log1pf(__expf(-fabsf(x)));
}

// ---------------------------------------------------------------------
// BPR loss via V_WMMA_F32_16X16X4_F32.
// One wave handles 16 batch items: A = ub rows (16x64), B = item rows as
// columns (64x16). Accumulate the 16x16 Gram tile over 16 K-steps of 4;
// the diagonal holds the per-item dot products. pos (i) and neg (j) tiles
// are computed in the same loop (2 WMMAs per K-step).
//
// Operand layout per ISA 7.12.2 (32-bit A 16x4): lane L (L<16) holds
// {K=4t, 4t+1} for M=L ; lane L (>=16) holds {K=4t+2, 4t+3} for M=L-16.
// B (4x16) mirrors with N in place of M -> both loads are float2 at
// offset 4t + 2*(L>>4) within the 64-float row.
// D diagonal: m<8 -> lane m, vgpr m ; m>=8 -> lane m+16, vgpr m-8.
// ---------------------------------------------------------------------
__global__ __launch_bounds__(256) void lightgcn_bpr_wmma(
    const float* __restrict__ ego,
    const int* __restrict__ u, const int* __restrict__ iidx,
    const int* __restrict__ jidx, float* __restrict__ out)
{
    const int lane = threadIdx.x & 31;
    const int wave = (blockIdx.x * blockDim.x + threadIdx.x) >> 5;
    const int base = wave << 4;          // 16 batch items per wave
    const int half = lane >> 4;
    const int lm   = lane & 15;
    const int b    = base + lm;

    const float* ub = ego + (long long)u[b] * EMB;
    const float* ip = ego + (long long)(N_USERS + iidx[b]) * EMB;
    const float* jn = ego + (long long)(N_USERS + jidx[b]) * EMB;

    v8f cpos = {0.f, 0.f, 0.f, 0.f, 0.f, 0.f, 0.f, 0.f};
    v8f cneg = {0.f, 0.f, 0.f, 0.f, 0.f, 0.f, 0.f, 0.f};

#pragma unroll
    for (int t = 0; t < 16; ++t) {
        const int off = 4 * t + 2 * half;
        v2f a  = *(const v2f*)(ub + off);
        v2f bp = *(const v2f*)(ip + off);
        v2f bn = *(const v2f*)(jn + off);
        cpos = __builtin_amdgcn_wmma_f32_16x16x4_f32(
            false, a, false, bp, (short)0, cpos, false, false);
        cneg = __builtin_amdgcn_wmma_f32_16x16x4_f32(
            false, a, false, bn, (short)0, cneg, false, false);
    }

    // extract diagonal element owned by this lane (if any)
    int r = (lane < 8) ? lane : ((lane >= 24) ? (lane - 24) : -1);
    float pos = 0.f, neg = 0.f;
#pragma unroll
    for (int q = 0; q < 8; ++q) {
        if (r == q) { pos = cpos[q]; neg = cneg[q]; }
    }

    float contrib = (r >= 0) ? (softplusf(-pos) + softplusf(neg)) : 0.f;

    // wave32 reduction
#pragma unroll
    for (int o = 16; o > 0; o >>= 1)
        contrib += __shfl_xor(contrib, o, 32);

    if (lane == 0)
        atomicAdd(out, contrib * (1.0f / (2.0f * (float)BATCH)));
}

// ---------------------------------------------------------------------
// launch
// ---------------------------------------------------------------------
extern "C" void kernel_launch(void* const* d_in, const int* in_sizes, int n_in,
                              void* d_out, int out_size, void* d_ws, size_t ws_size,
                              hipStream_t stream)
{
    const float* user_emb = (const float*)d_in[0];
    const float* item_emb = (const float*)d_in[1];
    const int*   row      = (const int*)d_in[2];
    const int*   col      = (const int*)d_in[3];
    const float* val      = (const float*)d_in[4];
    const int*   u        = (const int*)d_in[5];
    const int*   iidx     = (const int*)d_in[6];
    const int*   jidx     = (const int*)d_in[7];
    float*       out      = (float*)d_out;

    const size_t elems = (size_t)N_NODES * EMB;          // 12.8M floats
    const size_t bytes = elems * sizeof(float);          // 51.2 MB
    float* buf0 = (float*)d_ws;
    float* buf1 = buf0 + elems;

    const long long spmm_threads = NNZ * 32LL;           // wave per edge
    const int spmm_blocks = (int)((spmm_threads + 255) / 256);

    // layer 1: (user_emb | item_emb) -> buf0
    hipMemsetAsync(buf0, 0, bytes, stream);
    lightgcn_spmm<<<spmm_blocks, 256, 0, stream>>>(
        row, col, val, user_emb, item_emb, buf0, NNZ);

    // layer 2: buf0 -> buf1
    hipMemsetAsync(buf1, 0, bytes, stream);
    lightgcn_spmm<<<spmm_blocks, 256, 0, stream>>>(
        row, col, val, buf0, buf0 + (size_t)N_USERS * EMB, buf1, NNZ);

    // layer 3: buf1 -> buf0
    hipMemsetAsync(buf0, 0, bytes, stream);
    lightgcn_spmm<<<spmm_blocks, 256, 0, stream>>>(
        row, col, val, buf1, buf1 + (size_t)N_USERS * EMB, buf0, NNZ);

    // BPR loss: 16384 items, 16 per wave -> 1024 waves -> 128 blocks x 256
    hipMemsetAsync(out, 0, sizeof(float), stream);
    lightgcn_bpr_wmma<<<(BATCH / 16) * 32 / 256, 256, 0, stream>>>(
        buf0, u, iidx, jidx, out);
}